// self_attention_block_19370302505593
// MI455X (gfx1250) — compile-verified
//
#include <hip/hip_runtime.h>

// ---------------------------------------------------------------------------
// Masked entity self-attention block, fused single kernel for MI455X (gfx1250).
//   grid = 128 (one workgroup per batch row), block = 256 threads (8 wave32).
//   All intermediates live in LDS (~60 KB); attention GEMMs use
//   V_WMMA_F32_16X16X4_F32 (K=4 == head dim, f32 end-to-end, no downcast).
// ---------------------------------------------------------------------------

typedef __attribute__((ext_vector_type(2))) float v2f;
typedef __attribute__((ext_vector_type(8))) float v8f;

#define BS       128
#define NE       512
#define FDIM     6
#define HEADS    4
#define DPH      4
#define NEMBD    16
#define NTHREADS 256
#define NWAVES   8
#define QTILES   (NE / 16)   // 32
#define KTILES   (NE / 16)   // 32

static __device__ __forceinline__ v8f wmma16x16x4(v2f a, v2f b, v8f c) {
#if defined(__HIP_DEVICE_COMPILE__)
  // 8-arg pattern: (neg_a, A, neg_b, B, c_mod, C, reuse_a, reuse_b)
  return __builtin_amdgcn_wmma_f32_16x16x4_f32(false, a, false, b, (short)0, c,
                                               false, false);
#else
  (void)a; (void)b;
  return c;
#endif
}

static __device__ __forceinline__ void wave_fence() {
#if defined(__HIP_DEVICE_COMPILE__)
  __builtin_amdgcn_wave_barrier();
#endif
}

__global__ __launch_bounds__(NTHREADS, 1)
void attn_entity_block(const float* __restrict__ inp,    // [BS,NE,FDIM]
                       const float* __restrict__ maskG,  // [BS,NE]
                       const float* __restrict__ Wq,     // [NEMBD,FDIM]
                       const float* __restrict__ Wk,     // [NEMBD,FDIM]
                       const float* __restrict__ Wv,     // [NEMBD,FDIM]
                       const float* __restrict__ Wpost,  // [FDIM,NEMBD]
                       float* __restrict__ out)          // [BS,FDIM]
{
  __shared__ float xin[NE][FDIM];        // masked input, 12 KB
  __shared__ float attS[NE][17];         // attention output (16 cols used), 34 KB
  __shared__ float maskS[NE];
  __shared__ float biasS[NE];            // 1e9*(mask-1): -1e9 on masked keys
  __shared__ float wq[NEMBD][FDIM];
  __shared__ float wk[NEMBD][FDIM];
  __shared__ float wv[NEMBD][FDIM];
  __shared__ float wp[FDIM][NEMBD];
  __shared__ float stage[NWAVES][16][17];// per-wave P-tile relayout buffer
  __shared__ float red[16];              // block reduction accumulators

  const int  b    = blockIdx.x;
  const int  tid  = threadIdx.x;
  const int  lane = tid & 31;
  const int  wave = tid >> 5;
  const int  mrow = lane & 15;           // row (A) or column (B/C) index
  const bool hihalf = lane >= 16;
  const int  kk0  = hihalf ? 2 : 0;      // K slots owned by this half-wave

  // ---------------- phase 0: stage inputs into LDS ----------------
  for (int i = tid; i < NEMBD * FDIM; i += NTHREADS) {
    wq[i / FDIM][i % FDIM]   = Wq[i];
    wk[i / FDIM][i % FDIM]   = Wk[i];
    wv[i / FDIM][i % FDIM]   = Wv[i];
    wp[i / NEMBD][i % NEMBD] = Wpost[i];   // same element count (96)
  }
  for (int t = tid; t < NE; t += NTHREADS) {
    float m  = maskG[b * NE + t];
    maskS[t] = m;
    biasS[t] = 1e9f * (m - 1.0f);
  }
  for (int i = tid; i < NE * FDIM; i += NTHREADS) {
    int t = i / FDIM, f = i % FDIM;
    xin[t][f] = inp[(b * NE + t) * FDIM + f] * maskG[b * NE + t];
  }
  if (tid < 16) red[tid] = 0.0f;
  __syncthreads();

  // ---------------- phase 1: attention via WMMA ----------------
  // 128 (head, query-tile) pairs distributed over 8 waves.
  for (int pair = wave; pair < HEADS * QTILES; pair += NWAVES) {
    const int h  = pair & (HEADS - 1);
    const int qt = pair >> 2;

    // A operand: Q tile 16x4 (recomputed from xin: 12 FMA/lane, reused 32x)
    const int qrow = qt * 16 + mrow;
    float a0 = 0.0f, a1 = 0.0f;
#pragma unroll
    for (int f = 0; f < FDIM; ++f) {
      float xv = xin[qrow][f];
      a0 += xv * wq[h * DPH + kk0][f];
      a1 += xv * wq[h * DPH + kk0 + 1][f];
    }
    v2f Aq = {a0, a1};

    v8f   acc   = {0.f, 0.f, 0.f, 0.f, 0.f, 0.f, 0.f, 0.f}; // P@V accumulator
    float rs[8] = {0.f, 0.f, 0.f, 0.f, 0.f, 0.f, 0.f, 0.f}; // softmax row sums

    for (int kt = 0; kt < KTILES; ++kt) {
      // B operand: K tile 4x16 (column n = key kt*16+n)
      const int krow = kt * 16 + mrow;
      float b0 = 0.0f, b1 = 0.0f;
#pragma unroll
      for (int f = 0; f < FDIM; ++f) {
        float xv = xin[krow][f];
        b0 += xv * wk[h * DPH + kk0][f];
        b1 += xv * wk[h * DPH + kk0 + 1][f];
      }
      v2f Bk = {b0, b1};

      v8f zero = {0.f, 0.f, 0.f, 0.f, 0.f, 0.f, 0.f, 0.f};
      v8f lg   = wmma16x16x4(Aq, Bk, zero);  // 16x16 logits tile

      // softmax numerators (no max-subtraction: logits are O(1); masked keys
      // get -1e9 bias -> exp == 0; key 0 always unmasked so row sum > 0).
      const float kb = biasS[krow];          // this lane's column == key krow
#pragma unroll
      for (int v = 0; v < 8; ++v) {
        float p = __expf(lg[v] * 0.5f + kb); // scale = 1/sqrt(DPH) = 0.5
        rs[v] += p;
        stage[wave][v + (hihalf ? 8 : 0)][mrow] = p; // C-layout -> LDS
      }
      wave_fence(); // stage writes precede the A-layout reads below

      // P@V: split 16-key tile into 4 chunks of K=4, accumulate via WMMA.
#pragma unroll
      for (int k4 = 0; k4 < 4; ++k4) {
        v2f Pa = { stage[wave][mrow][k4 * 4 + kk0],
                   stage[wave][mrow][k4 * 4 + kk0 + 1] };
        // B operand: V chunk 4x16 (only N=0..3 valid = head dims)
        float vb0 = 0.0f, vb1 = 0.0f;
        if (mrow < DPH) {
          const int key0 = kt * 16 + k4 * 4 + kk0;
#pragma unroll
          for (int f = 0; f < FDIM; ++f) {
            float wvf = wv[h * DPH + mrow][f];
            vb0 += xin[key0][f]     * wvf;
            vb1 += xin[key0 + 1][f] * wvf;
          }
        }
        v2f Bv = {vb0, vb1};
        acc = wmma16x16x4(Pa, Bv, acc);
      }
      wave_fence(); // stage is reused next kt iteration
    }

    // reduce row sums across the 16 lanes of each half-wave
#pragma unroll
    for (int v = 0; v < 8; ++v) {
      float r = rs[v];
      r += __shfl_xor(r, 1, 32);
      r += __shfl_xor(r, 2, 32);
      r += __shfl_xor(r, 4, 32);
      r += __shfl_xor(r, 8, 32);
      rs[v] = r;
    }

    // write attention output; zero masked query rows (p * mask[:, :, None])
    if (mrow < DPH) {
#pragma unroll
      for (int v = 0; v < 8; ++v) {
        const int m = qt * 16 + v + (hihalf ? 8 : 0);
        attS[m][h * DPH + mrow] = (acc[v] / rs[v]) * maskS[m];
      }
    }
  }
  __syncthreads();

  // ---------------- phase 2: post-projection + residual + reductions -------
  // Masked rows of x are exactly 0 (xin==0, att==0), so the mask-corrected
  // normalization collapses to three scalars plus 6 per-feature masked sums:
  //   mu    = Sx / (N*F)
  //   Sx2   = Sx + (NE-N)*F*mu        (masked slots filled with mu)
  //   Sq2   = Sxx + (NE-N)*F*mu^2
  //   var   = (Sq2 - NE*F*m2^2) / (NE*F - 1),  m2 = Sx2/(NE*F)
  float pv[9];
#pragma unroll
  for (int i = 0; i < 9; ++i) pv[i] = 0.0f;

  for (int t = tid; t < NE; t += NTHREADS) {
    const float m = maskS[t];
#pragma unroll
    for (int f = 0; f < FDIM; ++f) {
      float s = xin[t][f];
#pragma unroll
      for (int e = 0; e < NEMBD; ++e) s += attS[t][e] * wp[f][e];
      pv[1]     += s;
      pv[2]     += s * s;
      pv[3 + f] += s * m;
    }
    pv[0] += m;
  }
#pragma unroll
  for (int i = 0; i < 9; ++i) {
    float r = pv[i];
    r += __shfl_xor(r, 1, 32);
    r += __shfl_xor(r, 2, 32);
    r += __shfl_xor(r, 4, 32);
    r += __shfl_xor(r, 8, 32);
    r += __shfl_xor(r, 16, 32);
    if (lane == 0) atomicAdd(&red[i], r);   // ds_add_f32
  }
  __syncthreads();

  // ---------------- phase 3: scalar epilogue + output ----------------
  if (tid < FDIM) {
    const float Nv  = red[0];
    const float sx  = red[1];
    const float sq  = red[2];
    const float nef = (float)(NE * FDIM);
    const float mu  = sx / (Nv * (float)FDIM);
    const float sx2 = sx + ((float)NE - Nv) * (float)FDIM * mu;
    const float m2  = sx2 / nef;
    const float sq2 = sq + ((float)NE - Nv) * (float)FDIM * mu * mu;
    const float var = (sq2 - nef * m2 * m2) / (nef - 1.0f);
    const float stdv =
        sqrtf(var) * sqrtf((nef - 1.0f) / ((float)FDIM * Nv - 1.0f));
    out[b * FDIM + tid] = (red[3 + tid] / Nv - mu) / (stdv + 1e-6f);
  }
}

extern "C" void kernel_launch(void* const* d_in, const int* in_sizes, int n_in,
                              void* d_out, int out_size, void* d_ws,
                              size_t ws_size, hipStream_t stream) {
  (void)in_sizes; (void)n_in; (void)out_size; (void)d_ws; (void)ws_size;
  const float* inp   = (const float*)d_in[0];
  const float* mask  = (const float*)d_in[1];
  const float* Wq    = (const float*)d_in[2];
  const float* Wk    = (const float*)d_in[3];
  const float* Wv    = (const float*)d_in[4];
  const float* Wpost = (const float*)d_in[5];
  float* out = (float*)d_out;
  attn_entity_block<<<dim3(BS), dim3(NTHREADS), 0, stream>>>(
      inp, mask, Wq, Wk, Wv, Wpost, out);
}